// CRF_70626442215505
// MI455X (gfx1250) — compile-verified
//
#include <hip/hip_runtime.h>
#include <stdint.h>

// ---------------------------------------------------------------------------
// Viterbi CRF decode for MI455X (gfx1250).
// Max-plus recursion is NOT WMMA-expressible (semiring (max,+) + argmax carry),
// so the CDNA5 paths used here are: 320KB-LDS-resident transition matrix AND
// full backpointer history, TDM async tensor_load_to_lds double-buffered
// emission streaming (TENSORcnt), wave32 ds_load_b128 broadcast/column reads.
// ---------------------------------------------------------------------------

typedef uint32_t u32;
typedef uint64_t u64;
typedef __attribute__((ext_vector_type(4))) u32 u32x4;
typedef __attribute__((ext_vector_type(8))) int i32x8;
typedef __attribute__((ext_vector_type(4))) int i32x4;

#define KK        128            // number of tags
#define TT        1024           // sequence length
#define CH        32             // emission timesteps per TDM chunk (16 KB)
#define NCHUNK    (TT / CH)

// ---- dynamic LDS layout (bytes) -------------------------------------------
#define OFF_TRANS  0
#define SZ_TRANS   (KK * KK * 4)                 // 65536: swizzled transitions
#define OFF_EMIS   (OFF_TRANS + SZ_TRANS)        // 65536
#define SZ_EMIS    (2 * CH * KK * 4)             // 32768: double-buffered chunks
#define OFF_HIST   (OFF_EMIS + SZ_EMIS)          // 98304
#define HIST_PITCH (TT + 4)                      // 1028: bank-conflict-free rows
#define SZ_HIST    (KK * HIST_PITCH)             // 131584: full backptr history
#define OFF_LABEL  (OFF_HIST + SZ_HIST)          // 229888
#define SZ_LABEL   (TT * 4)                      // 4096
#define OFF_SCORE  (OFF_LABEL + SZ_LABEL)        // 233984: ping-pong scores
#define SZ_SCORE   (2 * KK * 4)                  // 1024
#define OFF_REDV   (OFF_SCORE + SZ_SCORE)
#define OFF_REDI   (OFF_REDV + KK * 4)
#define LDS_BYTES  (OFF_REDI + KK * 4)           // ~236 KB (< 320 KB WGP LDS)

// ---- TDM 1-D tile load: global -> LDS, descriptor per CDNA5 ISA §8.3/8.4 ---
// clang-23 toolchain uses the 6-arg builtin:
//   (u32x4 group0, i32x8 group1, i32x4 group2, i32x4 group3, i32x8 extra, i32 cpol)
__device__ __forceinline__ void tdm_load_1d(u32 ldsOff, u64 gaddr, u32 nElemF32)
{
    u32x4 g0;
    g0.x = 0x1u;                                    // count=1, user, no gather
    g0.y = ldsOff;                                  // lds_addr [63:32]
    g0.z = (u32)(gaddr & 0xFFFFFFFFu);              // global_addr[31:0]
    g0.w = (u32)((gaddr >> 32) & 0x01FFFFFFu)       // global_addr[56:32]
         | (2u << 30);                              // type = 2 ("image")

    i32x8 g1;
    g1[0] = (int)(2u << 16);                        // data_size = 4 bytes
    g1[1] = (int)((nElemF32 & 0xFFFFu) << 16);      // tensor_dim0[15:0]
    g1[2] = (int)(((nElemF32 >> 16) & 0xFFFFu))     // tensor_dim0[31:16]
          | (1 << 16);                              // tensor_dim1 = 1
    g1[3] = (int)((nElemF32 & 0xFFFFu) << 16);      // tile_dim0 = nElem (<=65535)
    g1[4] = 1;                                      // tile_dim1 = 1, tile_dim2 = 0
    g1[5] = (int)nElemF32;                          // tensor_dim0_stride[31:0]
    g1[6] = 0;                                      // stride hi / dim1_stride lo
    g1[7] = 0;

    i32x4 z4 = {0, 0, 0, 0};                        // groups 2/3 unused (<=2D)
    i32x8 z8 = {0, 0, 0, 0, 0, 0, 0, 0};            // extra descriptor words
    __builtin_amdgcn_tensor_load_to_lds(g0, g1, z4, z4, z8, 0);
}

__global__ void __launch_bounds__(KK, 1)
viterbi_crf_kernel(const float* __restrict__ em,      // (B,T,K)
                   const int*   __restrict__ mask,    // (B,T)
                   const float* __restrict__ startT,  // (K)
                   const float* __restrict__ endT,    // (K)
                   const float* __restrict__ trans,   // (K,K)
                   float*       __restrict__ out)     // (B,T) labels
{
    extern __shared__ char smem[];
    float*         transL = (float*)(smem + OFF_TRANS);
    float*         emisL  = (float*)(smem + OFF_EMIS);
    unsigned char* hist   = (unsigned char*)(smem + OFF_HIST);
    int*           labelL = (int*)(smem + OFF_LABEL);
    float*         scoreL = (float*)(smem + OFF_SCORE);   // [2][KK]
    float*         redV   = (float*)(smem + OFF_REDV);
    int*           redI   = (int*)(smem + OFF_REDI);

    const int  j     = threadIdx.x;        // tag column owned by this lane
    const int  b     = blockIdx.x;         // batch element owned by this block
    const bool wave0 = (j < 32);

    // ---- transitions -> LDS, swizzled so 4 consecutive i for fixed j are
    //      contiguous: float index (i>>2)*4K + j*4 + (i&3)  => ds_load_b128
    for (int i = 0; i < KK; ++i) {
        float v = trans[i * KK + j];                       // coalesced
        transL[(i >> 2) * (4 * KK) + (j << 2) + (i & 3)] = v;
    }

    // ---- async-prefetch emission chunk 0 via TDM (wave 0 drives the DMA) ---
    const u64 emBase = (u64)(uintptr_t)(em + (u64)b * TT * KK);
    const u32 emLds  = (u32)(uintptr_t)emisL;              // low 32 = LDS offset
    if (wave0) {
        tdm_load_1d(emLds, emBase, CH * KK);
        __builtin_amdgcn_s_wait_tensorcnt(0);
    }
    __syncthreads();                                       // trans + chunk0 ready

    scoreL[j] = startT[j] + emisL[j];                      // t = 0 init
    if (wave0) {                                           // prefetch chunk 1
        tdm_load_1d(emLds + CH * KK * 4, emBase + (u64)CH * KK * 4, CH * KK);
    }
    __syncthreads();

    int p = 0;                                             // current score buf
    const float4* tr4 = (const float4*)transL;

    for (int c = 0; c < NCHUNK; ++c) {
        const float* eb   = emisL + (c & 1) * (CH * KK);
        const int    tBeg = (c == 0) ? 1 : c * CH;
        const int    tEnd = c * CH + CH;

        for (int t = tBeg; t < tEnd; ++t) {
            const float*  sc  = scoreL + p * KK;
            const float4* sc4 = (const float4*)sc;

            float best = -3.0e38f;
            int   bi   = 0;
            #pragma unroll
            for (int i4 = 0; i4 < KK / 4; ++i4) {
                float4 s  = sc4[i4];                // ds_load_b128 broadcast
                float4 tv = tr4[i4 * KK + j];       // ds_load_b128, no conflict
                float v0 = s.x + tv.x; if (v0 > best) { best = v0; bi = i4 * 4 + 0; }
                float v1 = s.y + tv.y; if (v1 > best) { best = v1; bi = i4 * 4 + 1; }
                float v2 = s.z + tv.z; if (v2 > best) { best = v2; bi = i4 * 4 + 2; }
                float v3 = s.w + tv.w; if (v3 > best) { best = v3; bi = i4 * 4 + 3; }
            }

            float nxt = best + eb[(t - c * CH) * KK + j];
            int   m   = mask[b * TT + t];                  // uniform -> s_load
            float ns  = (m > 0) ? nxt : sc[j];
            int   bp  = (m > 0) ? bi : 0;

            hist[j * HIST_PITCH + t] = (unsigned char)bp;  // ds_store_b8
            scoreL[(p ^ 1) * KK + j] = ns;
            __syncthreads();                               // one barrier/step
            p ^= 1;
        }

        if (c + 1 < NCHUNK) {
            if (wave0) __builtin_amdgcn_s_wait_tensorcnt(0); // chunk c+1 landed
            __syncthreads();                                 // publish to waves
            if (wave0 && (c + 2) < NCHUNK) {                 // refill freed buf
                tdm_load_1d(emLds + (u32)((c & 1) * CH * KK * 4),
                            emBase + (u64)(c + 2) * CH * KK * 4, CH * KK);
            }
        }
    }

    // ---- end-tag argmax + sequence length --------------------------------
    redV[j] = scoreL[p * KK + j] + endT[j];
    int msum = 0;
    for (int t = j; t < TT; t += KK) msum += (mask[b * TT + t] > 0) ? 1 : 0;
    labelL[j] = msum;                                      // temp reuse
    __syncthreads();

    if (j == 0) {
        float bv = redV[0]; int bt = 0;
        for (int i = 1; i < KK; ++i)                       // first-max tiebreak
            if (redV[i] > bv) { bv = redV[i]; bt = i; }
        int tot = 0;
        for (int i = 0; i < KK; ++i) tot += labelL[i];
        const int seqEnd = tot - 1;
        redI[0] = seqEnd;

        // ---- backtrace entirely from LDS-resident history ----------------
        int best = bt;
        labelL[seqEnd] = best;
        for (int t = seqEnd; t >= 1; --t) {
            best = hist[best * HIST_PITCH + t];            // ds_load_u8
            labelL[t - 1] = best;
        }
    }
    __syncthreads();

    const int seqEnd = redI[0];
    for (int t = j; t < TT; t += KK) {
        float o = (t <= seqEnd) ? (float)labelL[t] : -100.0f;  // PAD_TAG
        out[(u64)b * TT + t] = o;
    }
}

extern "C" void kernel_launch(void* const* d_in, const int* in_sizes, int n_in,
                              void* d_out, int out_size, void* d_ws, size_t ws_size,
                              hipStream_t stream)
{
    const float* em     = (const float*)d_in[0];
    const int*   mask   = (const int*)d_in[1];
    const float* startT = (const float*)d_in[2];
    const float* endT   = (const float*)d_in[3];
    const float* trans  = (const float*)d_in[4];
    float*       out    = (float*)d_out;

    const int B = in_sizes[1] / TT;                        // 256

    // Opt-in to >64KB dynamic LDS (no-op if already allowed).
    (void)hipFuncSetAttribute((const void*)viterbi_crf_kernel,
                              hipFuncAttributeMaxDynamicSharedMemorySize,
                              (int)LDS_BYTES);

    viterbi_crf_kernel<<<B, KK, LDS_BYTES, stream>>>(em, mask, startT, endT,
                                                     trans, out);
}